// GELU281_22677427323089
// MI455X (gfx1250) — compile-verified
//
#include <hip/hip_runtime.h>
#include <math.h>
#include <stdint.h>

// ---------------------------------------------------------------------------
// Gated-GELU (elementwise + per-row cosine-novelty gate) for MI455X / gfx1250.
//
// Memory-bound op: 268 MB of HBM traffic -> ~11.5us floor @ 23.3 TB/s.
// Structure: 1 block = 8 rows of D=2048. x rows are staged HBM -> LDS with
// gfx1250 async-to-LDS DMA (double buffered, tracked by ASYNCcnt), per-channel
// EMA constants live in registers for the whole block, per-row dot/norm uses
// wave32 shuffles + an 8-entry LDS tree, and the output is written with
// non-temporal B128 stores. Per-element tanh uses gfx1250's native
// V_TANH_F32 when the builtin is available (1 trans op instead of exp+rcp).
// ---------------------------------------------------------------------------

#define THREADS 256
#define DD      2048           // channel dim (specialized)
#define CPT     8              // channels per thread  (THREADS * CPT == DD)
#define NWAVES  (THREADS / 32) // wave32
#define RPB     8              // rows per block

typedef float v4f __attribute__((ext_vector_type(4)));

__device__ __forceinline__ float rcp_fast(float x) { return __builtin_amdgcn_rcpf(x); }
__device__ __forceinline__ float sqrt_fast(float x) { return __builtin_amdgcn_sqrtf(x); }

// Native v_tanh_f32 on gfx1250 if clang exposes it; otherwise
// tanh(x) = 1 - 2/(exp(2x)+1)  (exact at +-inf, v_exp_f32 + v_rcp_f32).
__device__ __forceinline__ float tanh_fast(float x) {
#if __has_builtin(__builtin_amdgcn_tanhf)
  return __builtin_amdgcn_tanhf(x);
#else
  float e = __expf(2.0f * x);
  return 1.0f - 2.0f * rcp_fast(e + 1.0f);
#endif
}
__device__ __forceinline__ float gelu_tanh(float x) {
  float u = 0.7978845608028654f * __builtin_fmaf(0.044715f * x, x * x, x);
  return 0.5f * x * (1.0f + tanh_fast(u));
}
__device__ __forceinline__ float softplus_f(float v) {
  return (v > 20.0f) ? v : __logf(1.0f + __expf(v));
}
__device__ __forceinline__ float clampf(float v, float lo, float hi) {
  return fminf(fmaxf(v, lo), hi);
}

// Issue 32 bytes (2 x B128) of global->LDS async DMA. offset: applies to both
// the LDS and global addresses (ISA 10.x async pseudocode), so one asm block
// covers the thread's 8 floats. Tracked by ASYNCcnt (+2 per call per wave).
__device__ __forceinline__ void async_row_chunk(uint32_t lds_off, const float* g) {
  asm volatile("global_load_async_to_lds_b128 %0, %1, off\n\t"
               "global_load_async_to_lds_b128 %0, %1, off offset:16"
               :: "v"(lds_off), "v"(g) : "memory");
}

// Sum-reduce two values over the block: wave32 xor-shuffle tree, then an
// 8-entry LDS combine that every thread reads (no second barrier needed).
__device__ __forceinline__ void block_reduce2(float& a, float& b,
                                              float (*sred)[2], int wave, int lane) {
  #pragma unroll
  for (int off = 16; off > 0; off >>= 1) {
    a += __shfl_xor(a, off, 32);
    b += __shfl_xor(b, off, 32);
  }
  if (lane == 0) { sred[wave][0] = a; sred[wave][1] = b; }
  __syncthreads();
  float ra = 0.0f, rb = 0.0f;
  #pragma unroll
  for (int w = 0; w < NWAVES; ++w) { ra += sred[w][0]; rb += sred[w][1]; }
  a = ra; b = rb;
}

__global__ __launch_bounds__(THREADS)
void gelu_adagate_kernel(const float* __restrict__ x,
                         const float* __restrict__ ema_mean,
                         const float* __restrict__ ema_sq,
                         const float* __restrict__ ema_out_mean,
                         const float* __restrict__ ema_out_sq,
                         const float* __restrict__ ema_out_dir,
                         const float* __restrict__ p_log_tau,
                         const float* __restrict__ p_log_beta_up,
                         const float* __restrict__ p_log_beta_dn,
                         const float* __restrict__ p_log_gamma,
                         const float* __restrict__ p_log_beta_out,
                         const float* __restrict__ p_log_gamma_out,
                         float* __restrict__ out,
                         int nrows) {
  __shared__ alignas(16) float s_x[2][DD];     // ping-pong row buffers (16 KiB)
  __shared__ float s_red[NWAVES][2];

  const int tid  = threadIdx.x;
  const int wave = tid >> 5;
  const int lane = tid & 31;
  const int d0   = tid * CPT;

  const int row0 = blockIdx.x * RPB;
  if (row0 >= nrows) return;                    // uniform per block
  const int rows_here = (nrows - row0 < RPB) ? (nrows - row0) : RPB;

  // Kick off the DMA for row 0 immediately; all parameter setup below
  // (scalar softplus chain + per-channel stats) overlaps with it.
  {
    uint32_t l = (uint32_t)(uintptr_t)&s_x[0][d0];   // low 32 bits == LDS addr
    async_row_chunk(l, x + (size_t)row0 * DD + d0);
  }

  // Scalar gate parameters (uniform -> compiler scalarizes to v_s_exp/v_s_log).
  const float tau       = __expf(p_log_tau[0]);
  const float beta_up   = softplus_f(p_log_beta_up[0]);
  const float beta_dn   = softplus_f(p_log_beta_dn[0]);
  const float gamma     = softplus_f(p_log_gamma[0]);
  const float beta_out  = softplus_f(p_log_beta_out[0]);
  const float gamma_out = softplus_f(p_log_gamma_out[0]);

  // Per-thread channel constants, held in registers for the whole block.
  float m_in[CPT], i_in[CPT], m_out[CPT], i_out[CPT], dirv[CPT];
  float ssd = 0.0f;
  #pragma unroll
  for (int j = 0; j < CPT; ++j) {
    const int d = d0 + j;
    m_in[j] = ema_mean[d];
    float vi = fmaxf(ema_sq[d] - m_in[j] * m_in[j], 1e-4f);
    i_in[j]  = rcp_fast(sqrt_fast(vi) + 1e-5f);
    m_out[j] = ema_out_mean[d];
    float vo = fmaxf(ema_out_sq[d] - m_out[j] * m_out[j], 1e-4f);
    i_out[j] = rcp_fast(sqrt_fast(vo) + 1e-5f);
    dirv[j]  = ema_out_dir[d];
    ssd = __builtin_fmaf(dirv[j], dirv[j], ssd);
  }
  float zero = 0.0f;
  block_reduce2(ssd, zero, s_red, wave, lane);          // ||ema_out_dir||^2
  const float inv_dir_n = rcp_fast(fmaxf(sqrt_fast(ssd), 1e-12f));

  for (int i = 0; i < rows_here; ++i) {
    const int buf = i & 1;

    // Prefetch row i+1 into the other buffer, then retire row i's DMA.
    if (i + 1 < rows_here) {
      uint32_t l = (uint32_t)(uintptr_t)&s_x[buf ^ 1][d0];
      async_row_chunk(l, x + (size_t)(row0 + i + 1) * DD + d0);
      asm volatile("s_wait_asynccnt 2" ::: "memory");   // oldest 2 (row i) done
    } else {
      asm volatile("s_wait_asynccnt 0" ::: "memory");
    }
    __syncthreads();  // row i visible block-wide; also fences last row's s_red

    // Row i from LDS (ds_load_b128 x2), GELU + reduction partials.
    v4f xa = *(const v4f*)&s_x[buf][d0];
    v4f xb = *(const v4f*)&s_x[buf][d0 + 4];
    float xs[CPT] = {xa.x, xa.y, xa.z, xa.w, xb.x, xb.y, xb.z, xb.w};

    float o[CPT];
    float dot = 0.0f, ss = 0.0f;
    #pragma unroll
    for (int j = 0; j < CPT; ++j) {
      float g = gelu_tanh(xs[j]);
      o[j] = g;
      dot = __builtin_fmaf(g, dirv[j], dot);
      ss  = __builtin_fmaf(g, g, ss);
    }

    // Barrier inside also releases s_x[buf] for the i+2 prefetch.
    block_reduce2(dot, ss, s_red, wave, lane);

    const float inv_on = rcp_fast(fmaxf(sqrt_fast(ss), 1e-12f));
    const float cosv   = clampf(dot * inv_on * inv_dir_n, -1.0f, 1.0f);
    const float gcos   = __expf(-tau * cosv);

    float res[CPT];
    #pragma unroll
    for (int j = 0; j < CPT; ++j) {
      float zi = (xs[j] - m_in[j]) * i_in[j];
      float t  = tanh_fast(gamma * zi);
      float gi = clampf(1.0f + beta_up * fmaxf(t, 0.0f)
                             - beta_dn * fmaxf(-t, 0.0f), 0.05f, 8.0f);
      float zo = (o[j] - m_out[j]) * i_out[j];
      float go = clampf(__builtin_fmaf(beta_out, tanh_fast(gamma_out * zo), 1.0f),
                        0.1f, 5.0f);
      float gr = clampf(gi * go * gcos, 0.01f, 20.0f);
      res[j] = o[j] * gr;
    }
    v4f r0 = {res[0], res[1], res[2], res[3]};
    v4f r1 = {res[4], res[5], res[6], res[7]};
    float* op = out + (size_t)(row0 + i) * DD + d0;
    __builtin_nontemporal_store(r0, (v4f*)op);       // streaming output: NT
    __builtin_nontemporal_store(r1, (v4f*)(op + 4));
  }
}

extern "C" void kernel_launch(void* const* d_in, const int* in_sizes, int n_in,
                              void* d_out, int out_size, void* d_ws, size_t ws_size,
                              hipStream_t stream) {
  (void)n_in; (void)out_size; (void)d_ws; (void)ws_size;
  const float* x            = (const float*)d_in[0];
  const float* ema_mean     = (const float*)d_in[1];
  const float* ema_sq       = (const float*)d_in[2];
  const float* ema_out_mean = (const float*)d_in[3];
  const float* ema_out_sq   = (const float*)d_in[4];
  const float* ema_out_dir  = (const float*)d_in[5];
  const float* log_tau      = (const float*)d_in[6];
  const float* log_beta_up  = (const float*)d_in[7];
  const float* log_beta_dn  = (const float*)d_in[8];
  const float* log_gamma    = (const float*)d_in[9];
  const float* log_beta_out = (const float*)d_in[10];
  const float* log_gamma_out= (const float*)d_in[11];

  const int n = in_sizes[0];
  const int D = in_sizes[1];
  if (D != DD || n <= 0) return;      // kernel is specialized for D == 2048
  const int nrows  = n / D;
  const int blocks = (nrows + RPB - 1) / RPB;

  gelu_adagate_kernel<<<blocks, THREADS, 0, stream>>>(
      x, ema_mean, ema_sq, ema_out_mean, ema_out_sq, ema_out_dir,
      log_tau, log_beta_up, log_beta_dn, log_gamma, log_beta_out, log_gamma_out,
      (float*)d_out, nrows);
}